// GaussPSF_58935541236087
// MI455X (gfx1250) — compile-verified
//
#include <hip/hip_runtime.h>
#include <cstdint>

#define KSZ    7
#define PADR   3
#define IMG_H  768
#define IMG_W  768
#define NCH    3
#define TW     64
#define TH     16
#define HALO_W (TW + 2 * PADR)   // 70
#define HALO_H (TH + 2 * PADR)   // 22
#define SSTR   72                // padded LDS row stride (floats)
#define PIXEL_SIZE 5.6e-06f
#define SCALE      3.0f
// -log2(e)/2
#define NEG_HALF_LOG2E -0.72134752044448170367f

typedef __attribute__((address_space(1))) int glb_i32;
typedef __attribute__((address_space(3))) int lds_i32;

// ---- CDNA5 async global->LDS copy (ASYNCcnt path) -------------------------
__device__ __forceinline__ void async_ld_b32(const float* g, float* l) {
#if defined(__AMDGCN__) && __has_builtin(__builtin_amdgcn_global_load_async_to_lds_b32)
  __builtin_amdgcn_global_load_async_to_lds_b32(
      (glb_i32*)(g), (lds_i32*)(l), 0, 0);
#elif defined(__AMDGCN__)
  unsigned lofs = (unsigned)(unsigned long long)(lds_i32*)l;
  asm volatile("global_load_async_to_lds_b32 %0, %1, off"
               :: "v"(lofs), "v"(g) : "memory");
#else
  *l = *g;  // host-pass parse only
#endif
}

__device__ __forceinline__ void wait_async_zero() {
#if defined(__AMDGCN__) && __has_builtin(__builtin_amdgcn_s_wait_asynccnt)
  __builtin_amdgcn_s_wait_asynccnt(0);
#elif defined(__AMDGCN__)
  asm volatile("s_wait_asynccnt 0" ::: "memory");
#endif
}

__global__ __launch_bounds__(256) void GaussPSF_58935541236087_kernel(
    const float* __restrict__ image,
    const float* __restrict__ depth,
    const float* __restrict__ focal_depth,
    const float* __restrict__ Nv,
    const float* __restrict__ focal_len,
    float* __restrict__ out) {
  __shared__ float s_a2[HALO_H * SSTR];
  __shared__ float s_img[NCH][HALO_H * SSTR];

  const int tid = threadIdx.x;
  const int b   = blockIdx.z;
  const int x0  = blockIdx.x * TW;
  const int y0  = blockIdx.y * TH;

  const float fd = focal_depth[b];
  const float FL = focal_len[b];
  const float K1 = fabsf((FL * FL / Nv[b]) / (fd - FL)) * (SCALE / PIXEL_SIZE);

  const size_t plane = (size_t)IMG_H * IMG_W;
  const float* imgb = image + (size_t)b * NCH * plane;
  const float* depb = depth + (size_t)b * plane;

  // ---- Stage halo tile: sigma plane computed from depth; 3 image planes
  // ---- copied HBM->LDS with async DMA (no VGPR round trip).
  for (int idx = tid; idx < HALO_H * HALO_W; idx += 256) {
    const int hy = idx / HALO_W;
    const int hx = idx - hy * HALO_W;
    const int gy = y0 - PADR + hy;
    const int gx = x0 - PADR + hx;
    const int lo = hy * SSTR + hx;
    if (gy >= 0 && gy < IMG_H && gx >= 0 && gx < IMG_W) {
      const size_t g = (size_t)gy * IMG_W + gx;
      const float d = depb[g];
      const float c = K1 * fabsf((d - fd) / d);
      const float w = fminf(fmaxf(c, 1.0f), (float)KSZ);
      s_a2[lo] = NEG_HALF_LOG2E / (w * w);
      async_ld_b32(imgb + g,              &s_img[0][lo]);
      async_ld_b32(imgb + plane + g,      &s_img[1][lo]);
      async_ld_b32(imgb + 2 * plane + g,  &s_img[2][lo]);
    } else {
      // padded region: weight 0 (v = exp2(-3e38 * r2) == 0, r2 >= 1), image 0
      s_a2[lo] = -3.0e38f;
      s_img[0][lo] = 0.0f;
      s_img[1][lo] = 0.0f;
      s_img[2][lo] = 0.0f;
    }
  }
  wait_async_zero();
  __syncthreads();

  // ---- Compute: each thread owns 4 vertically adjacent pixels of the tile.
  const int px  = tid & (TW - 1);    // 0..63
  const int py0 = (tid >> 6) * 4;    // 0,4,8,12

  float acc0[4] = {0.f, 0.f, 0.f, 0.f};
  float acc1[4] = {0.f, 0.f, 0.f, 0.f};
  float acc2[4] = {0.f, 0.f, 0.f, 0.f};
  float wsum[4] = {0.f, 0.f, 0.f, 0.f};

#pragma unroll
  for (int k = 0; k < 4; ++k) {
#pragma unroll
    for (int dy = -PADR; dy <= PADR; ++dy) {
      const int row = (py0 + k + dy + PADR) * SSTR;
#pragma unroll
      for (int dx = -PADR; dx <= PADR; ++dx) {
        const int o = row + (px + dx + PADR);
        const float r2 = (float)(dy * dy + dx * dx);
        const float v = __builtin_amdgcn_exp2f(s_a2[o] * r2);  // v_exp_f32
        wsum[k] += v;
        acc0[k] += v * s_img[0][o];
        acc1[k] += v * s_img[1][o];
        acc2[k] += v * s_img[2][o];
      }
    }
  }

  float* outb = out + (size_t)b * NCH * plane;
#pragma unroll
  for (int k = 0; k < 4; ++k) {
    const size_t g = (size_t)(y0 + py0 + k) * IMG_W + (x0 + px);
    outb[g]             = acc0[k] / wsum[k];
    outb[plane + g]     = acc1[k] / wsum[k];
    outb[2 * plane + g] = acc2[k] / wsum[k];
  }
}

extern "C" void kernel_launch(void* const* d_in, const int* in_sizes, int n_in,
                              void* d_out, int out_size, void* d_ws, size_t ws_size,
                              hipStream_t stream) {
  (void)n_in; (void)out_size; (void)d_ws; (void)ws_size;
  const float* image = (const float*)d_in[0];
  const float* depth = (const float*)d_in[1];
  const float* fd    = (const float*)d_in[2];
  const float* Nv    = (const float*)d_in[3];
  const float* fl    = (const float*)d_in[4];
  float* out = (float*)d_out;

  const int B = in_sizes[2];  // focal_depth has one entry per batch
  dim3 grid(IMG_W / TW, IMG_H / TH, B);
  GaussPSF_58935541236087_kernel<<<grid, dim3(256), 0, stream>>>(
      image, depth, fd, Nv, fl, out);
}